// AttentionConv_1580547969859
// MI455X (gfx1250) — compile-verified
//
#include <hip/hip_runtime.h>

typedef __attribute__((ext_vector_type(2))) float v2f;
typedef __attribute__((ext_vector_type(8))) float v8f;

#define BATCH 4
#define CIN   64
#define HH    56
#define WW    56
#define COUT  64
#define NPIX  (HH * WW)            // 3136
#define PIX_TILES (NPIX / 16)      // 196
#define OT    (COUT / 16)          // 4
#define PLANE ((size_t)BATCH * COUT * NPIX)  // 802816 elements per projection

// ---------------------------------------------------------------------------
// Kernel 1: q/k/v 1x1-conv as WMMA GEMM.  D(16x16) = W(16x64) * X(64x16) per
// wave, K swept in steps of 4 with V_WMMA_F32_16X16X4_F32 (exact fp32).
// ---------------------------------------------------------------------------
__global__ __launch_bounds__(256)
void qkv_gemm_wmma(const float* __restrict__ x,
                   const float* __restrict__ qw,
                   const float* __restrict__ kw,
                   const float* __restrict__ vw,
                   float* __restrict__ ws)
{
    const int wid  = (blockIdx.x * blockDim.x + threadIdx.x) >> 5; // global wave id
    const int lane = threadIdx.x & 31;

    // wid in [0, 196*4*4*3): decompose (pixel tile, cout tile, batch, projection)
    int t  = wid;
    const int pt    = t % PIX_TILES;  t /= PIX_TILES;
    const int ot    = t % OT;         t /= OT;
    const int b     = t % BATCH;      t /= BATCH;
    const int which = t;              // 0 = q, 1 = k, 2 = v

    const float* Wm = (which == 0) ? qw : (which == 1) ? kw : vw;
    float* outp = ws + (size_t)which * PLANE;

    const int n  = lane & 15;   // A row m == B col n == lane%16
    const int hi = lane >> 4;   // lane half selects K sub-pairs (ISA 7.12.2)

    const float* wrow = Wm + (size_t)(ot * 16 + n) * CIN;              // A: W[m, :]
    const float* xcol = x + (size_t)b * CIN * NPIX + (size_t)(pt * 16 + n); // B: X[:, n]

    v8f acc = {};
#pragma unroll
    for (int kb = 0; kb < CIN; kb += 4) {
        const int c = kb + hi * 2;
        v2f a, bx;
        a.x  = wrow[c];
        a.y  = wrow[c + 1];
        bx.x = xcol[(size_t)c * NPIX];
        bx.y = xcol[(size_t)(c + 1) * NPIX];
        // 8 args: (neg_a, A, neg_b, B, c_mod, C, reuse_a, reuse_b)
        acc = __builtin_amdgcn_wmma_f32_16x16x4_f32(
            false, a, false, bx, (short)0, acc, false, false);
    }

    // D layout: VGPR r -> (M = r + 8*hi, N = lane&15)
    float* orow = outp + ((size_t)(b * COUT + ot * 16) * NPIX) + (size_t)(pt * 16 + n);
#pragma unroll
    for (int r = 0; r < 8; ++r) {
        const int m = r + hi * 8;
        orow[(size_t)m * NPIX] = acc[r];
    }
}

// ---------------------------------------------------------------------------
// Kernel 2: per-element 7x7 window softmax-attention.
// logit(i,j) = q * (k[h+i-3, w+j-3] + rel);   OOB tap -> k contribution 0
// (rel still added, matching reference which adds rel after unfold of the
// zero-padded tensor), v OOB -> 0.
// ---------------------------------------------------------------------------
__global__ __launch_bounds__(256)
void attn_window(const float* __restrict__ ws,
                 const float* __restrict__ rel_h,
                 const float* __restrict__ rel_w,
                 float* __restrict__ out)
{
    const int idx = blockIdx.x * 256 + threadIdx.x;      // < 802816
    const int wq = idx % WW;
    const int hq = (idx / WW) % HH;
    const int c  = (idx / NPIX) % COUT;

    const float* qb = ws;
    const float* kb = ws + PLANE;
    const float* vb = ws + 2 * PLANE;

    const float qv   = qb[idx];
    const int   base = idx - (hq * WW + wq);   // start of this (b,c) plane
    const bool  useH = (c < 32);

    float rel[7];
#pragma unroll
    for (int t = 0; t < 7; ++t)
        rel[t] = useH ? rel_h[c * 7 + t] : rel_w[(c - 32) * 7 + t];

    // pass 1: running max of logits
    float mx = -3.402823466e+38f;
#pragma unroll
    for (int i = 0; i < 7; ++i) {
        const int hh = hq + i - 3;
        const bool hin = ((unsigned)hh < (unsigned)HH);
#pragma unroll
        for (int j = 0; j < 7; ++j) {
            const int wj = wq + j - 3;
            const bool inb = hin && ((unsigned)wj < (unsigned)WW);
            float kv = inb ? kb[base + hh * WW + wj] : 0.0f;
            kv += useH ? rel[i] : rel[j];
            mx = fmaxf(mx, qv * kv);
        }
    }

    // pass 2: exp-sum and weighted v accumulation
    float sum = 0.0f, dot = 0.0f;
#pragma unroll
    for (int i = 0; i < 7; ++i) {
        const int hh = hq + i - 3;
        const bool hin = ((unsigned)hh < (unsigned)HH);
#pragma unroll
        for (int j = 0; j < 7; ++j) {
            const int wj = wq + j - 3;
            const bool inb = hin && ((unsigned)wj < (unsigned)WW);
            const int off = base + hh * WW + wj;
            float kv = inb ? kb[off] : 0.0f;
            kv += useH ? rel[i] : rel[j];
            const float p = __expf(qv * kv - mx);
            sum += p;
            if (inb) dot += p * vb[off];
        }
    }
    out[idx] = dot / sum;
}

// ---------------------------------------------------------------------------
extern "C" void kernel_launch(void* const* d_in, const int* in_sizes, int n_in,
                              void* d_out, int out_size, void* d_ws, size_t ws_size,
                              hipStream_t stream) {
    const float* x   = (const float*)d_in[0];
    const float* q_w = (const float*)d_in[1];
    const float* k_w = (const float*)d_in[2];
    const float* v_w = (const float*)d_in[3];
    const float* r_h = (const float*)d_in[4];
    const float* r_w = (const float*)d_in[5];
    float* out = (float*)d_out;
    float* ws  = (float*)d_ws;   // needs 3 * 802816 * 4 B ~= 9.6 MB

    // 196*4*4*3 = 9408 waves -> 1176 blocks of 8 waves
    qkv_gemm_wmma<<<1176, 256, 0, stream>>>(x, q_w, k_w, v_w, ws);
    // 802816 elements -> 3136 blocks of 256 threads
    attn_window<<<3136, 256, 0, stream>>>(ws, r_h, r_w, out);
}